// RecursiveMamba130M_29566554866430
// MI455X (gfx1250) — compile-verified
//
#include <hip/hip_runtime.h>
#include <math.h>

typedef __attribute__((ext_vector_type(2))) float v2f;
typedef __attribute__((ext_vector_type(8))) float v8f;

#define BB 2
#define LL 1024
#define DD 768
#define CC 2
#define SS 16
#define BL (BB * LL)   // 2048 rows
#define CD (CC * DD)   // 1536

// ---------------------------------------------------------------------------
// Kernel 0: precompute cos/sin of A_theta (C*D*S = 24576 elements, trivial)
// ---------------------------------------------------------------------------
__global__ void k_sincos(const float* __restrict__ theta,
                         float* __restrict__ ct, float* __restrict__ st, int n) {
    int i = blockIdx.x * blockDim.x + threadIdx.x;
    if (i < n) {
        float s, c;
        sincosf(theta[i], &s, &c);
        ct[i] = c;
        st[i] = s;
    }
}

// ---------------------------------------------------------------------------
// WMMA f32 GEMM:  Out[M][N] = X[M][K] @ W[N][K]^T   (both row-major)
// K, N are compile-time so all strides fold into immediate load/store offsets
// (24-bit byte offsets; 16*K*4 <= 96KB here) -> inner loop is one pointer
// bump + clause loads + WMMAs.
// Each wave computes a 16x64 output strip = 4 WMMA tiles sharing one A
// fragment per K-step (load:WMMA ratio 1.25 instead of 2.0; A traffic /4).
// A-frag (16x4 f32, 2 VGPR): lanes 0-15 -> K={k,k+1}, lanes 16-31 -> K={k+2,k+3}
// B-frag mirrors A; C/D: VGPR i -> row (i + 8*half), col = lane&15.
// Requires N % 64 == 0 (holds: 1536, 768). Guard is wave-uniform, so EXEC
// stays all-ones inside every wave that reaches the WMMA ops.
// ---------------------------------------------------------------------------
template<int K, int N>
__global__ void k_gemm_nt_wmma(const float* __restrict__ X,
                               const float* __restrict__ W,
                               float* __restrict__ Out,
                               int M) {
    const int lane    = threadIdx.x & 31;
    const int wave    = threadIdx.x >> 5;
    const int stripsN = N >> 6;                       // 64-wide column strips
    const int strip   = blockIdx.x * (blockDim.x >> 5) + wave;
    if (strip >= (M >> 4) * stripsN) return;

    const int tm   = strip / stripsN;
    const int tn   = strip % stripsN;
    const int half = lane >> 4;      // 0: K pair {0,1}, 1: K pair {2,3}
    const int r    = lane & 15;

    // Base pointers already shifted by the lane's K-pair (2*half floats).
    const float* __restrict__ xp = X + (size_t)(tm * 16 + r) * K + 2 * half;
    const float* __restrict__ wp = W + (size_t)(tn * 64 + r) * K + 2 * half;

    v8f a0 = {}, a1 = {}, a2 = {}, a3 = {};
#pragma unroll 4
    for (int k0 = 0; k0 < K; k0 += 4) {
        v2f a  = *(const v2f*)(xp + k0);
        v2f b0 = *(const v2f*)(wp + k0);
        v2f b1 = *(const v2f*)(wp + 16 * K + k0);     // immediate offsets
        v2f b2 = *(const v2f*)(wp + 32 * K + k0);
        v2f b3 = *(const v2f*)(wp + 48 * K + k0);
        a0 = __builtin_amdgcn_wmma_f32_16x16x4_f32(false, a, false, b0, (short)0, a0, false, false);
        a1 = __builtin_amdgcn_wmma_f32_16x16x4_f32(false, a, false, b1, (short)0, a1, false, false);
        a2 = __builtin_amdgcn_wmma_f32_16x16x4_f32(false, a, false, b2, (short)0, a2, false, false);
        a3 = __builtin_amdgcn_wmma_f32_16x16x4_f32(false, a, false, b3, (short)0, a3, false, false);
    }

    float* __restrict__ orow = Out + (size_t)(tm * 16 + 8 * half) * N + tn * 64 + r;
#pragma unroll
    for (int i = 0; i < 8; ++i) {
        orow[i * N +  0] = a0[i];                     // immediate offsets
        orow[i * N + 16] = a1[i];
        orow[i * N + 32] = a2[i];
        orow[i * N + 48] = a3[i];
    }
}

// ---------------------------------------------------------------------------
// State rotation + readout (bandwidth-bound phase, ~805 MB HBM traffic).
// 4 lanes per (b,l,c,d); each lane owns one float4 of the S axis.
//  new_real = cos*rs - sin*is + Br*x ; new_imag = sin*rs + cos*is + Bi*x
//  y(b,l,c,d) = sum_s (Cr*new_real - Ci*new_imag)   via 2x shfl_xor in-quad
// ---------------------------------------------------------------------------
__global__ void k_state(const float* __restrict__ xin,
                        const float* __restrict__ rs,  const float* __restrict__ im,
                        const float* __restrict__ ct,  const float* __restrict__ st,
                        const float* __restrict__ Br,  const float* __restrict__ Bi,
                        const float* __restrict__ Cr,  const float* __restrict__ Ci,
                        float* __restrict__ nr, float* __restrict__ ni,
                        float* __restrict__ y) {
    const long long g   = (long long)blockIdx.x * blockDim.x + threadIdx.x;
    const long long q   = g >> 2;                 // (bl*C*D + c*D + d)
    const int       sub = (int)(g & 3);           // which float4 of S=16
    const int       cd  = (int)(q % CD);          // c*D + d
    const long long soff = q * SS + sub * 4;      // state offset
    const int       coff = cd * SS + sub * 4;     // coefficient offset

    const float xv = xin[q];

    float4 r0 = *(const float4*)(rs + soff);
    float4 i0 = *(const float4*)(im + soff);
    float4 c0 = *(const float4*)(ct + coff);
    float4 s0 = *(const float4*)(st + coff);
    float4 br = *(const float4*)(Br + coff);
    float4 bi = *(const float4*)(Bi + coff);
    float4 cr = *(const float4*)(Cr + coff);
    float4 ci = *(const float4*)(Ci + coff);

    float4 R, I;
    R.x = fmaf(c0.x, r0.x, fmaf(-s0.x, i0.x, br.x * xv));
    R.y = fmaf(c0.y, r0.y, fmaf(-s0.y, i0.y, br.y * xv));
    R.z = fmaf(c0.z, r0.z, fmaf(-s0.z, i0.z, br.z * xv));
    R.w = fmaf(c0.w, r0.w, fmaf(-s0.w, i0.w, br.w * xv));
    I.x = fmaf(s0.x, r0.x, fmaf( c0.x, i0.x, bi.x * xv));
    I.y = fmaf(s0.y, r0.y, fmaf( c0.y, i0.y, bi.y * xv));
    I.z = fmaf(s0.z, r0.z, fmaf( c0.z, i0.z, bi.z * xv));
    I.w = fmaf(s0.w, r0.w, fmaf( c0.w, i0.w, bi.w * xv));

    *(float4*)(nr + soff) = R;
    *(float4*)(ni + soff) = I;

    float part = cr.x * R.x - ci.x * I.x
               + cr.y * R.y - ci.y * I.y
               + cr.z * R.z - ci.z * I.z
               + cr.w * R.w - ci.w * I.w;
    part += __shfl_xor(part, 1, 32);
    part += __shfl_xor(part, 2, 32);
    if (sub == 0) y[q] = part;
}

// ---------------------------------------------------------------------------
// RMSNorm + residual: out = x + h * rsqrt(mean(h^2) + eps) * w  (row = 768)
// ---------------------------------------------------------------------------
__global__ void k_rmsnorm_res(const float* __restrict__ h,
                              const float* __restrict__ x,
                              const float* __restrict__ w,
                              float* __restrict__ out) {
    const int row = blockIdx.x;             // 0..BL-1
    const int tid = threadIdx.x;            // 256 threads, 3 elems each
    const float* __restrict__ hr = h + (size_t)row * DD;
    const float* __restrict__ xr = x + (size_t)row * DD;
    float* __restrict__       orr = out + (size_t)row * DD;

    float v0 = hr[tid], v1 = hr[tid + 256], v2 = hr[tid + 512];
    float ss = v0 * v0 + v1 * v1 + v2 * v2;
    for (int o = 16; o > 0; o >>= 1) ss += __shfl_down(ss, o, 32);

    __shared__ float red[8];
    __shared__ float total;
    if ((tid & 31) == 0) red[tid >> 5] = ss;
    __syncthreads();
    if (tid == 0) {
        float t = 0.f;
#pragma unroll
        for (int i = 0; i < 8; ++i) t += red[i];
        total = t;
    }
    __syncthreads();

    const float inv = rsqrtf(total * (1.0f / (float)DD) + 1.1920928955078125e-07f);
    orr[tid]       = xr[tid]       + v0 * inv * w[tid];
    orr[tid + 256] = xr[tid + 256] + v1 * inv * w[tid + 256];
    orr[tid + 512] = xr[tid + 512] + v2 * inv * w[tid + 512];
}

// ---------------------------------------------------------------------------
extern "C" void kernel_launch(void* const* d_in, const int* in_sizes, int n_in,
                              void* d_out, int out_size, void* d_ws, size_t ws_size,
                              hipStream_t stream) {
    (void)in_sizes; (void)n_in; (void)out_size; (void)ws_size;

    const float* x     = (const float*)d_in[0];
    const float* rs    = (const float*)d_in[1];
    const float* im    = (const float*)d_in[2];
    const float* W_in  = (const float*)d_in[3];
    const float* A_th  = (const float*)d_in[4];
    const float* B_r   = (const float*)d_in[5];
    const float* B_i   = (const float*)d_in[6];
    const float* C_r   = (const float*)d_in[7];
    const float* C_i   = (const float*)d_in[8];
    const float* W_out = (const float*)d_in[9];
    const float* nw    = (const float*)d_in[10];

    float* out  = (float*)d_out;                                   // (B,L,D)
    float* nr   = out + (size_t)BL * DD;                           // new_real
    float* ni   = nr  + (size_t)BL * CD * SS;                      // new_imag

    float* ws  = (float*)d_ws;
    float* xin = ws;                          // BL*CD
    float* y   = xin + (size_t)BL * CD;       // BL*CD
    float* hh  = y   + (size_t)BL * CD;       // BL*DD
    float* ct  = hh  + (size_t)BL * DD;       // CD*SS
    float* st  = ct  + (size_t)CD * SS;       // CD*SS

    // 0) cos/sin tables
    {
        int n = CD * SS;
        k_sincos<<<(n + 255) / 256, 256, 0, stream>>>(A_th, ct, st, n);
    }
    // 1) x_in = x @ W_in.T    (M=2048, N=1536, K=768): 128x24 strips
    {
        int strips = (BL / 16) * (CD / 64);
        k_gemm_nt_wmma<DD, CD><<<(strips + 7) / 8, 256, 0, stream>>>(x, W_in, xin, BL);
    }
    // 2) state rotation + readout (writes new_real/new_imag into d_out)
    {
        long long threads = (long long)BL * CD * 4;  // 4 lanes per (b,l,c,d)
        k_state<<<(unsigned)(threads / 256), 256, 0, stream>>>(
            xin, rs, im, ct, st, B_r, B_i, C_r, C_i, nr, ni, y);
    }
    // 3) h = y @ W_out.T      (M=2048, N=768, K=1536): 128x12 strips
    {
        int strips = (BL / 16) * (DD / 64);
        k_gemm_nt_wmma<CD, DD><<<(strips + 7) / 8, 256, 0, stream>>>(y, W_out, hh, BL);
    }
    // 4) out = x + rmsnorm(h) * w
    k_rmsnorm_res<<<BL, 256, 0, stream>>>(hh, x, nw, out);
}